// Attention_7679401525457
// MI455X (gfx1250) — compile-verified
//
#include <hip/hip_runtime.h>
#include <hip/hip_bf16.h>

// Problem constants (from reference): B=32, S=1024, D=1024, L=2
constexpr int Bdim = 32;
constexpr int Sdim = 1024;
constexpr int Ddim = 1024;

typedef unsigned int u32;
typedef u32 v4u __attribute__((ext_vector_type(4)));
typedef int  v8i __attribute__((ext_vector_type(8)));
typedef int  v4i __attribute__((ext_vector_type(4)));

// ---------------------------------------------------------------------------
// gfx1250-specific helpers
// ---------------------------------------------------------------------------

// Hardware tanh (gfx1250 VOP1 transcendental). Pure -> non-volatile asm so the
// scheduler can batch surrounding loads. v_nop covers the TRANS-op hazard.
__device__ __forceinline__ float fast_tanh(float x) {
    float y;
    asm("v_tanh_f32 %0, %1\n\tv_nop" : "=v"(y) : "v"(x));
    return y;
}

// Async global->LDS copy: each active lane moves 16 bytes.
__device__ __forceinline__ void async_ld_b128(unsigned lds_addr, const void* gaddr) {
    asm volatile("global_load_async_to_lds_b128 %0, %1, off"
                 :: "v"(lds_addr), "v"(gaddr) : "memory");
}
__device__ __forceinline__ void async_ld_b128_off512(unsigned lds_addr, const void* gaddr) {
    asm volatile("global_load_async_to_lds_b128 %0, %1, off offset:512"
                 :: "v"(lds_addr), "v"(gaddr) : "memory");
}
__device__ __forceinline__ void wait_async_le2() {
    asm volatile("s_wait_asynccnt 0x2" ::: "memory");
}
__device__ __forceinline__ void wait_async_0() {
    asm volatile("s_wait_asynccnt 0x0" ::: "memory");
}

// Generic pointer to a __shared__ object: low 32 bits are the LDS byte offset.
__device__ __forceinline__ unsigned lds_offset_of(const void* p) {
    return (unsigned)(unsigned long long)p;
}

__device__ __forceinline__ float wave_sum(float v) {
    #pragma unroll
    for (int off = 16; off > 0; off >>= 1) v += __shfl_xor(v, off, 32);
    return v;
}
__device__ __forceinline__ float wave_max(float v) {
    #pragma unroll
    for (int off = 16; off > 0; off >>= 1) v = fmaxf(v, __shfl_xor(v, off, 32));
    return v;
}

#if __has_builtin(__builtin_amdgcn_tensor_load_to_lds)
#define HAVE_TDM 1
// Issue one TDM descriptor: 2D tile, 16 rows x 256 f32 (1KB/row), row stride
// 1024 f32 (4KB) in global memory, packed contiguously into LDS at lds_addr.
// D# layout per CDNA5 ISA ch.8 (CDNA5 ISA).
// This toolchain exposes the 6-arg builtin:
//   (v4u group0, v8i group1, v4i group2, v4i group3, v8i group4, i32 cpol)
__device__ __forceinline__ void tdm_load_tile_16x256(unsigned lds_addr, const void* gaddr) {
    const unsigned long long ga = (unsigned long long)gaddr;
    // group0: [1:0]=count=1 | lds_addr | global_addr[56:0] | type=2 (bits127:126)
    const v4u g0 = { 1u,
                     lds_addr,
                     (u32)(ga & 0xffffffffu),
                     ((u32)(ga >> 32) & 0x01ffffffu) | 0x80000000u };
    // group1:
    //  w0: workgroup_mask=0, data_size=2 (4B) at bits[17:16]
    //  tensor_dim0 = 1<<20 (bits 79:48), tensor_dim1 = 1<<20 (bits 111:80)
    //  tile_dim0   = 256  (bits 127:112)
    //  tile_dim1   = 16   (bits 143:128), tile_dim2 = 0 (unused)
    //  tensor_dim0_stride = 1024 (bits 207:160); dim1 stride = 0 (unused, 2D)
    const v8i g1 = { (int)0x00020000u,          // data_size = 4 bytes
                     0,                          // tensor_dim0[15:0] << 16 = 0
                     (int)0x00000010u,           // tensor_dim0[31:16]=0x10
                     (int)0x01000010u,           // tensor_dim1[31:16]=0x10 | tile_dim0=256
                     16,                         // tile_dim1 = 16 rows
                     1024,                       // tensor_dim0_stride = 1024 elems
                     0, 0 };
    const v4i gz4 = { 0, 0, 0, 0 };              // groups 2/3: dims 2..4 unused
    const v8i gz8 = { 0, 0, 0, 0, 0, 0, 0, 0 };  // extra group (unused)
    __builtin_amdgcn_tensor_load_to_lds(g0, g1, gz4, gz4, gz8, 0);
}
#endif

// ---------------------------------------------------------------------------
// Kernel 0: tz[b] = tanh(z[b,:]) . W2 + bias   (z = decoder_hidden[L-1])
// ---------------------------------------------------------------------------
__global__ void __launch_bounds__(256)
tz_kernel(const float* __restrict__ dec, const float* __restrict__ W,
          const float* __restrict__ bias, float* __restrict__ tz, int zoff) {
    const int b = blockIdx.x;
    const float* z  = dec + zoff + (size_t)b * Ddim;
    const float* w2 = W + Ddim;             // W[d:, 0]
    float acc = 0.f;
    #pragma unroll
    for (int k = 0; k < Ddim / 256; ++k) {
        const int i = k * 256 + threadIdx.x;
        acc = fmaf(fast_tanh(z[i]), w2[i], acc);
    }
    acc = wave_sum(acc);
    __shared__ float red[8];
    const int lane = threadIdx.x & 31, wv = threadIdx.x >> 5;
    if (lane == 0) red[wv] = acc;
    __syncthreads();
    if (threadIdx.x == 0) {
        float s = 0.f;
        #pragma unroll
        for (int i = 0; i < 8; ++i) s += red[i];
        tz[b] = s + bias[0];
    }
}

// ---------------------------------------------------------------------------
// Kernel 1: scores[b,s] = relu( sum_d tanh(enc[b,s,d])*W1[d] + tz[b] )
// One wave per (b,s) row; row streamed through LDS in 1KB chunks with
// double-buffered async global->LDS copies.
// ---------------------------------------------------------------------------
__global__ void __launch_bounds__(256)
scores_kernel(const float* __restrict__ enc, const float* __restrict__ W,
              const float* __restrict__ tz, float* __restrict__ scores) {
    __shared__ float smem[8 * 512];                 // 8 waves * (2 x 256 floats)
    const int lane = threadIdx.x & 31;
    const int wave = threadIdx.x >> 5;
    const int row  = blockIdx.x * 8 + wave;         // row = b*S + s
    const int b    = row >> 10;                     // S = 1024
    const float* rowp = enc + (size_t)row * Ddim;
    const unsigned ldsbase = lds_offset_of(&smem[wave * 512]);

    // Prologue: chunk 0 into buffer 0 (2 x 512B)
    {
        const char* g = (const char*)rowp + lane * 16;
        async_ld_b128(ldsbase + lane * 16, g);
        async_ld_b128_off512(ldsbase + lane * 16, g);
    }

    float acc = 0.f;
    #pragma unroll
    for (int c = 0; c < 4; ++c) {
        if (c < 3) {
            const int nb = (c + 1) & 1;
            const char* g = (const char*)(rowp + (c + 1) * 256) + lane * 16;
            const unsigned la = ldsbase + nb * 1024 + lane * 16;
            async_ld_b128(la, g);
            async_ld_b128_off512(la, g);
            wait_async_le2();
        } else {
            wait_async_0();
        }
        const float4* bufl = (const float4*)&smem[wave * 512 + (c & 1) * 256];
        const float4  e0 = bufl[lane];          // d = c*256 + lane*4 + [0..3]
        const float4  e1 = bufl[32 + lane];     // d = c*256 + 128 + lane*4 + [0..3]
        const float4* wp = (const float4*)(W + c * 256);
        const float4  w0 = wp[lane];
        const float4  w1 = wp[32 + lane];
        acc = fmaf(fast_tanh(e0.x), w0.x, acc);
        acc = fmaf(fast_tanh(e0.y), w0.y, acc);
        acc = fmaf(fast_tanh(e0.z), w0.z, acc);
        acc = fmaf(fast_tanh(e0.w), w0.w, acc);
        acc = fmaf(fast_tanh(e1.x), w1.x, acc);
        acc = fmaf(fast_tanh(e1.y), w1.y, acc);
        acc = fmaf(fast_tanh(e1.z), w1.z, acc);
        acc = fmaf(fast_tanh(e1.w), w1.w, acc);
    }

    acc = wave_sum(acc);
    if (lane == 0) scores[row] = fmaxf(acc + tz[b], 0.f);
}

// ---------------------------------------------------------------------------
// Kernel 2: in-place softmax over s (S=1024) per batch.
// ---------------------------------------------------------------------------
__global__ void __launch_bounds__(1024)
softmax_kernel(float* __restrict__ sc) {
    const int b = blockIdx.x;
    const int lane = threadIdx.x & 31, wv = threadIdx.x >> 5;
    __shared__ float red[32];
    const float v = sc[b * Sdim + threadIdx.x];

    float m = wave_max(v);
    if (lane == 0) red[wv] = m;
    __syncthreads();
    m = wave_max(red[lane]);

    const float e = __expf(v - m);

    float s = wave_sum(e);
    __syncthreads();
    if (lane == 0) red[wv] = s;
    __syncthreads();
    s = wave_sum(red[lane]);

    sc[b * Sdim + threadIdx.x] = e / s;
}

// ---------------------------------------------------------------------------
// Kernel 3: partial[b, sc, d] = sum_{s in chunk sc} alpha[b,s] * enc[b,s,d]
// grid = (B, D/256, S/256); block = 256.
// TDM variant: 16-row x 1KB tiles DMA'd into LDS, double-buffered with
// TENSORcnt; all 8 waves consume after a barrier.
// ---------------------------------------------------------------------------
#ifdef HAVE_TDM
__global__ void __launch_bounds__(256)
attn_partial_tdm(const float* __restrict__ enc, const float* __restrict__ alphas,
                 float* __restrict__ partial) {
    __shared__ float tile[2 * 16 * 256];            // 2 x 16KB
    __shared__ float a[256];
    const int b  = blockIdx.x;
    const int dt = blockIdx.y;
    const int sc = blockIdx.z;
    const int t  = threadIdx.x;
    a[t] = alphas[b * Sdim + sc * 256 + t];
    const float* gbase = enc + ((size_t)b * Sdim + (size_t)sc * 256) * Ddim + dt * 256;

    if (threadIdx.x < 32)                          // wave 0 drives the TDM
        tdm_load_tile_16x256(lds_offset_of(&tile[0]), gbase);

    float acc = 0.f;
    for (int step = 0; step < 16; ++step) {
        if (threadIdx.x < 32) {
            if (step < 15) {
                tdm_load_tile_16x256(lds_offset_of(&tile[((step + 1) & 1) * 4096]),
                                     gbase + (size_t)(step + 1) * 16 * Ddim);
                __builtin_amdgcn_s_wait_tensorcnt(1);   // previous tile done
            } else {
                __builtin_amdgcn_s_wait_tensorcnt(0);
            }
        }
        __syncthreads();                            // tile[step&1] ready for all waves
        const float* buf = &tile[(step & 1) * 4096];
        const float* ap  = &a[step * 16];
        #pragma unroll
        for (int r = 0; r < 16; ++r)
            acc = fmaf(ap[r], buf[r * 256 + t], acc);
        __syncthreads();                            // everyone done before overwrite
    }
    partial[((size_t)(b * 4 + sc)) * Ddim + dt * 256 + t] = acc;
}
#endif

// Fallback / reference variant (plain coalesced loads).
__global__ void __launch_bounds__(256)
attn_partial_kernel(const float* __restrict__ enc, const float* __restrict__ alphas,
                    float* __restrict__ partial) {
    const int b  = blockIdx.x;
    const int dt = blockIdx.y;
    const int sc = blockIdx.z;
    const int t  = threadIdx.x;
    __shared__ float a[256];
    a[t] = alphas[b * Sdim + sc * 256 + t];
    __syncthreads();
    const float* base = enc + ((size_t)b * Sdim + (size_t)sc * 256) * Ddim + dt * 256 + t;
    float acc = 0.f;
    #pragma unroll 4
    for (int i = 0; i < 256; ++i)
        acc = fmaf(a[i], base[(size_t)i * Ddim], acc);
    partial[((size_t)(b * 4 + sc)) * Ddim + dt * 256 + t] = acc;
}

// ---------------------------------------------------------------------------
// Kernel 4: out[b,d] = sum over the 4 s-chunk partials (deterministic).
// ---------------------------------------------------------------------------
__global__ void __launch_bounds__(256)
attn_reduce_kernel(const float* __restrict__ partial, float* __restrict__ out) {
    const int idx = blockIdx.x * 256 + threadIdx.x;   // b*D + d
    const int b = idx >> 10;
    const int d = idx & 1023;
    const float* p = partial + (size_t)(b * 4) * Ddim + d;
    out[idx] = (p[0] + p[(size_t)Ddim]) + (p[(size_t)2 * Ddim] + p[(size_t)3 * Ddim]);
}

// ---------------------------------------------------------------------------
// Launch
// ---------------------------------------------------------------------------
extern "C" void kernel_launch(void* const* d_in, const int* in_sizes, int n_in,
                              void* d_out, int out_size, void* d_ws, size_t ws_size,
                              hipStream_t stream) {
    const float* enc  = (const float*)d_in[0];   // (B,S,D) f32
    const float* dec  = (const float*)d_in[1];   // (L,B,D) f32
    const float* W    = (const float*)d_in[2];   // (2D,1)  f32
    const float* bias = (const float*)d_in[3];   // (1,)    f32
    float* out = (float*)d_out;                  // (B,D)   f32

    float* tz      = (float*)d_ws;
    float* alphas  = tz + 64;
    float* partial = (float*)((char*)d_ws + 192 * 1024);

    const int zoff = in_sizes[1] - Bdim * Ddim;  // last layer of decoder_hidden

    tz_kernel<<<dim3(Bdim), dim3(256), 0, stream>>>(dec, W, bias, tz, zoff);
    scores_kernel<<<dim3((Bdim * Sdim) / 8), dim3(256), 0, stream>>>(enc, W, tz, alphas);
    softmax_kernel<<<dim3(Bdim), dim3(1024), 0, stream>>>(alphas);
#ifdef HAVE_TDM
    attn_partial_tdm<<<dim3(Bdim, Ddim / 256, Sdim / 256), dim3(256), 0, stream>>>(enc, alphas, partial);
#else
    attn_partial_kernel<<<dim3(Bdim, Ddim / 256, Sdim / 256), dim3(256), 0, stream>>>(enc, alphas, partial);
#endif
    attn_reduce_kernel<<<dim3((Bdim * Ddim) / 256), dim3(256), 0, stream>>>(partial, out);
}